// pointBase_conv_738734375498
// MI455X (gfx1250) — compile-verified
//
#include <hip/hip_runtime.h>
#include <math.h>

// ---------------------------------------------------------------------------
// MI455X (gfx1250) implementation. All GEMM-shaped math goes through
// V_WMMA_F32_16X16X4_F32 (wave32, A/B = 2 VGPR f32 frags, C/D = 8 VGPR f32).
// Pipeline is L2-resident (~25MB); kernels are fused around that.
// ---------------------------------------------------------------------------

typedef float v2f __attribute__((ext_vector_type(2)));
typedef float v8f __attribute__((ext_vector_type(8)));

#define WMMA_F32X4(A, Bm, C) \
  __builtin_amdgcn_wmma_f32_16x16x4_f32(false, (A), false, (Bm), (short)0, (C), false, false)

// problem sizes
#define NB 16
#define NP 4096
#define NS 1024
#define NK 24
#define NR (NS * NK)   // rows per batch for conv2 GEMM = 24576
#define NCO 64
#define DSTRIDE 516    // 516 % 64 == 4 -> distinct LDS banks per row

// ---------------------------------------------------------------------------
// kernel 0: zero the accumulator regions of the workspace
// ---------------------------------------------------------------------------
__global__ __launch_bounds__(256) void k_zero(float* a, int na, float* b, int nb,
                                              float* c, int nc) {
  int i = blockIdx.x * blockDim.x + threadIdx.x;
  if (i < na) a[i] = 0.0f;
  if (i < nb) b[i] = 0.0f;
  if (i < nc) c[i] = 0.0f;
}

// ---------------------------------------------------------------------------
// kernel 1: fused distance (WMMA f32 16x16x4) + top-24 selection + gather +
// group mean/center + per-block sum of squared centered values.
// grid = (S/16, B), block = 1 wave (32).
// Each half-wave maintains a sorted top-24 list for even/odd columns of its
// row (conflict-free LDS banks); the halves are merged at the end with
// top_k-compatible index tie-breaking.
// ---------------------------------------------------------------------------
__global__ __launch_bounds__(32) void k_knn(const float* __restrict__ xyz,
                                            const float* __restrict__ anc,
                                            float* __restrict__ grp,
                                            float* __restrict__ ctr,
                                            float* __restrict__ stdPart) {
  __shared__ float distL[16 * DSTRIDE];
  __shared__ float qqL[16];
  __shared__ float lval[32 * NK];
  __shared__ int   lidx[32 * NK];
  __shared__ float mval[16 * NK];
  __shared__ int   midx[16 * NK];
  __shared__ float red[16];

  const int st = blockIdx.x;          // s-tile 0..63
  const int b  = blockIdx.y;
  const int lane = threadIdx.x;
  const int lm = lane & 15, hl = lane >> 4;
  const int s0 = st * 16;
  const float* xb = xyz + b * 3 * NP;

  // A fragment: rows = 16 query points, K = {x,y,z,0}. Unconditional loads,
  // lane-select (no exec-mask divergence).
  const float* qrow = anc + ((2 * NB + b) * NS + (s0 + lm)) * 3;
  float q0 = qrow[0], q1 = qrow[1], q2 = qrow[2];
  v2f afrag;
  afrag.x = hl ? q2   : q0;
  afrag.y = hl ? 0.0f : q1;
  if (lane < 16) qqL[lane] = q0 * q0 + q1 * q1 + q2 * q2;
  for (int k = 0; k < NK; k++) {
    lval[lane * NK + k] = 3.4e38f;
    lidx[lane * NK + k] = 0x7fffffff;
  }
  __syncthreads();
  float qqr[8];
#pragma unroll
  for (int r = 0; r < 8; r++) qqr[r] = qqL[r + 8 * hl];

  for (int n0 = 0; n0 < NP; n0 += 512) {
    // 32 WMMA tiles of q·p; the lane owning column lm also computes that
    // column's squared norm locally (no pp phase, no duplicate loads).
    for (int t = 0; t < 32; t++) {
      int col = n0 + t * 16 + lm;
      float p0 = xb[col], p1 = xb[NP + col], p2 = xb[2 * NP + col];
      v2f bfrag;
      bfrag.x = hl ? p2   : p0;
      bfrag.y = hl ? 0.0f : p1;
      float pp = p0 * p0 + p1 * p1 + p2 * p2;
      v8f acc = {};
      acc = WMMA_F32X4(afrag, bfrag, acc);
#pragma unroll
      for (int r = 0; r < 8; r++)
        distL[(r + 8 * hl) * DSTRIDE + t * 16 + lm] = qqr[r] + pp - 2.0f * acc[r];
    }
    __syncthreads();
    // all 32 lanes select: lane (lm,hl) scans columns n = 2*j + hl of row lm
    {
      float* lv = lval + lane * NK;
      int*   li = lidx + lane * NK;
      const float* dr = distL + lm * DSTRIDE;
      float cur = lv[NK - 1];
      for (int j = 0; j < 256; j++) {
        int n = 2 * j + hl;
        float d = dr[n];
        if (d < cur) {
          int jj = NK - 1;
          while (jj > 0 && lv[jj - 1] > d) { lv[jj] = lv[jj - 1]; li[jj] = li[jj - 1]; jj--; }
          lv[jj] = d; li[jj] = n0 + n;
          cur = lv[NK - 1];
        }
      }
    }
    __syncthreads();
  }

  // merge the two sorted half-lists per row (ties -> lower index, like top_k)
  if (lane < 16) {
    const float* av = lval + lm * NK;        const int* ai = lidx + lm * NK;
    const float* bv = lval + (lm + 16) * NK; const int* bi = lidx + (lm + 16) * NK;
    int ia = 0, ib = 0;
    for (int k = 0; k < NK; k++) {
      float va = av[ia], vb = bv[ib];
      int   xa = ai[ia], xbi = bi[ib];
      bool takeA = (va < vb) || (va == vb && xa < xbi);
      mval[lm * NK + k] = takeA ? va : vb;
      midx[lm * NK + k] = takeA ? xa : xbi;
      if (takeA) ia++; else ib++;
    }
  }
  __syncthreads();

  // gather neighbors, group mean/center, local sum of squares
  if (lane < 16) {
    const int* li = midx + lane * NK;
    float mx = 0.f, my = 0.f, mz = 0.f;
    for (int k = 0; k < NK; k++) {
      int id = li[k];
      mx += xb[id]; my += xb[NP + id]; mz += xb[2 * NP + id];
    }
    mx *= (1.0f / NK); my *= (1.0f / NK); mz *= (1.0f / NK);
    int base = ((b * NS + s0 + lane) * NK) * 3;
    float ss = 0.f;
    for (int k = 0; k < NK; k++) {
      int id = li[k];
      float p0 = xb[id], p1 = xb[NP + id], p2 = xb[2 * NP + id];
      float c0 = p0 - mx, c1 = p1 - my, c2 = p2 - mz;
      grp[base + k * 3 + 0] = p0; grp[base + k * 3 + 1] = p1; grp[base + k * 3 + 2] = p2;
      ctr[base + k * 3 + 0] = c0; ctr[base + k * 3 + 1] = c1; ctr[base + k * 3 + 2] = c2;
      ss += c0 * c0 + c1 * c1 + c2 * c2;
    }
    red[lane] = ss;
  }
  __syncthreads();
  if (lane == 0) {
    float s = 0.f;
    for (int i = 0; i < 16; i++) s += red[i];
    stdPart[b * 64 + st] = s;   // deterministic per-block partial
  }
}

// ---------------------------------------------------------------------------
// kernel 2: per-batch std finalize (unbiased over S*K*3 flattened, mean==0)
// ---------------------------------------------------------------------------
__global__ __launch_bounds__(32) void k_std(const float* __restrict__ sp,
                                            float* __restrict__ invstd) {
  int b = threadIdx.x;
  if (b < NB) {
    float s = 0.f;
    for (int i = 0; i < 64; i++) s += sp[b * 64 + i];   // fixed order
    float st = sqrtf(s / (float)(NS * NK * 3 - 1));
    invstd[b] = 1.0f / (st + 1e-5f);
  }
}

// helper: build the 4x2 B fragments of conv2_w (transposed, K padded 6->8)
__device__ __forceinline__ void load_w2_frags(const float* __restrict__ w2,
                                              int lm, int hl, v2f wB[4][2]) {
#pragma unroll
  for (int ot = 0; ot < 4; ot++) {
    int o = ot * 16 + lm;
#pragma unroll
    for (int ks = 0; ks < 2; ks++) {
      int k0 = ks * 4 + 2 * hl;
      wB[ot][ks].x = (k0     < 6) ? w2[o * 6 + k0]     : 0.0f;
      wB[ot][ks].y = (k0 + 1 < 6) ? w2[o * 6 + k0 + 1] : 0.0f;
    }
  }
}

// helper: A fragments (feat row = [grouped(3) | normalized(3)] padded to 8)
__device__ __forceinline__ void load_feat_frags(const float* __restrict__ gp,
                                                const float* __restrict__ cp,
                                                float inv, float a0, float a1, float a2,
                                                float e0, float e1, float e2,
                                                int hl, v2f& A0, v2f& A1) {
  float g0 = gp[0], g1 = gp[1], g2 = gp[2];
  float n0 = a0 * cp[0] * inv + e0;
  float n1 = a1 * cp[1] * inv + e1;
  float n2 = a2 * cp[2] * inv + e2;
  A0.x = hl ? g2  : g0;  A0.y = hl ? n0  : g1;
  A1.x = hl ? 0.f : n1;  A1.y = hl ? 0.f : n2;
}

// ---------------------------------------------------------------------------
// kernel 3: conv2 GEMM (stats pass): accumulate per-channel sum/sumsq of
// y = feat @ W2^T + b2 over all B*S*K rows. grid = (96, B), 1 wave.
// ---------------------------------------------------------------------------
__global__ __launch_bounds__(32) void k_c2stats(const float* __restrict__ grp,
                                                const float* __restrict__ ctr,
                                                const float* __restrict__ invstd,
                                                const float* __restrict__ al,
                                                const float* __restrict__ be,
                                                const float* __restrict__ w2,
                                                const float* __restrict__ b2,
                                                float* __restrict__ sum,
                                                float* __restrict__ sq) {
  __shared__ float ls[64], lq[64];
  int b = blockIdx.y, blk = blockIdx.x;
  int lane = threadIdx.x, lm = lane & 15, hl = lane >> 4;
  float inv = invstd[b];
  v2f wB[4][2];
  load_w2_frags(w2, lm, hl, wB);
  float a0 = al[0], a1 = al[1], a2 = al[2], e0 = be[0], e1 = be[1], e2 = be[2];
  float ps[4] = {0, 0, 0, 0}, pq[4] = {0, 0, 0, 0};
  float bias[4];
#pragma unroll
  for (int ot = 0; ot < 4; ot++) bias[ot] = b2[ot * 16 + lm];

  for (int tt = 0; tt < 16; tt++) {
    int row0 = b * NR + (blk * 16 + tt) * 16;
    const float* gp = grp + (row0 + lm) * 3;
    const float* cp = ctr + (row0 + lm) * 3;
    v2f A0, A1;
    load_feat_frags(gp, cp, inv, a0, a1, a2, e0, e1, e2, hl, A0, A1);
#pragma unroll
    for (int ot = 0; ot < 4; ot++) {
      v8f acc = {};
      acc = WMMA_F32X4(A0, wB[ot][0], acc);
      acc = WMMA_F32X4(A1, wB[ot][1], acc);
#pragma unroll
      for (int r = 0; r < 8; r++) {
        float y = acc[r] + bias[ot];
        ps[ot] += y; pq[ot] += y * y;
      }
    }
  }
  for (int i = lane; i < 64; i += 32) { ls[i] = 0.f; lq[i] = 0.f; }
  __syncthreads();
#pragma unroll
  for (int ot = 0; ot < 4; ot++) {
    atomicAdd(&ls[ot * 16 + lm], ps[ot]);
    atomicAdd(&lq[ot * 16 + lm], pq[ot]);
  }
  __syncthreads();
  for (int i = lane; i < 64; i += 32) {
    atomicAdd(&sum[i], ls[i]);
    atomicAdd(&sq[i],  lq[i]);
  }
}

// ---------------------------------------------------------------------------
// kernel 4/7: BN finalize -> fused scale/shift per channel
// ---------------------------------------------------------------------------
__global__ __launch_bounds__(64) void k_bnfin(const float* __restrict__ sum,
                                              const float* __restrict__ sq,
                                              const float* __restrict__ g,
                                              const float* __restrict__ bta,
                                              float* __restrict__ sc,
                                              float* __restrict__ sh, float M) {
  int o = threadIdx.x;
  float mean = sum[o] / M;
  float var  = sq[o] / M - mean * mean;
  float s = g[o] * rsqrtf(var + 1e-5f);
  sc[o] = s;
  sh[o] = bta[o] - s * mean;
}

// ---------------------------------------------------------------------------
// kernel 5: conv2 GEMM (apply pass): BN + ReLU + max over K -> x[B,64,S].
// grid = (S/32, B), block = 128 (4 waves). Max-over-K via int-bits atomicMax
// (valid: post-ReLU values are >= 0).
// ---------------------------------------------------------------------------
__global__ __launch_bounds__(128) void k_c2apply(const float* __restrict__ grp,
                                                 const float* __restrict__ ctr,
                                                 const float* __restrict__ invstd,
                                                 const float* __restrict__ al,
                                                 const float* __restrict__ be,
                                                 const float* __restrict__ w2,
                                                 const float* __restrict__ b2,
                                                 const float* __restrict__ sc,
                                                 const float* __restrict__ sh,
                                                 float* __restrict__ x) {
  __shared__ int xs[32 * 64];
  int b = blockIdx.y;
  int s0 = blockIdx.x * 32;
  int tid = threadIdx.x, wave = tid >> 5, lane = tid & 31;
  int lm = lane & 15, hl = lane >> 4;
  for (int i = tid; i < 32 * 64; i += 128) xs[i] = 0;
  float inv = invstd[b];
  v2f wB[4][2];
  load_w2_frags(w2, lm, hl, wB);
  float a0 = al[0], a1 = al[1], a2 = al[2], e0 = be[0], e1 = be[1], e2 = be[2];
  float bias[4], bsc[4], bsh[4];
#pragma unroll
  for (int ot = 0; ot < 4; ot++) {
    int o = ot * 16 + lm;
    bias[ot] = b2[o]; bsc[ot] = sc[o]; bsh[ot] = sh[o];
  }
  __syncthreads();
  int rowbase = s0 * NK;                       // row within batch
  for (int tt = wave; tt < 48; tt += 4) {      // 48 row-tiles cover 32 s-groups
    int row0 = rowbase + tt * 16;
    const float* gp = grp + (b * NR + row0 + lm) * 3;
    const float* cp = ctr + (b * NR + row0 + lm) * 3;
    v2f A0, A1;
    load_feat_frags(gp, cp, inv, a0, a1, a2, e0, e1, e2, hl, A0, A1);
#pragma unroll
    for (int ot = 0; ot < 4; ot++) {
      v8f acc = {};
      acc = WMMA_F32X4(A0, wB[ot][0], acc);
      acc = WMMA_F32X4(A1, wB[ot][1], acc);
#pragma unroll
      for (int r = 0; r < 8; r++) {
        int gr = row0 + r + 8 * hl;            // feat row -> (s,k)
        int sl = gr / NK - s0;
        float y = bsc[ot] * (acc[r] + bias[ot]) + bsh[ot];
        y = y > 0.f ? y : 0.f;
        atomicMax(&xs[sl * 64 + ot * 16 + lm], __float_as_int(y));
      }
    }
  }
  __syncthreads();
  for (int i = tid; i < 32 * 64; i += 128) {
    int sl = i >> 6, o = i & 63;
    x[(b * 64 + o) * NS + s0 + sl] = __int_as_float(xs[sl * 64 + o]);
  }
}

// ---------------------------------------------------------------------------
// kernel 6: Conv1d(64,64,k=3,pad=1) over S via WMMA (M=o, N=s, K=192) +
// per-channel BN1 stat accumulation. grid = (S/16, B), 1 wave.
// ---------------------------------------------------------------------------
__global__ __launch_bounds__(32) void k_conv1(const float* __restrict__ x,
                                              const float* __restrict__ w1,
                                              const float* __restrict__ b1,
                                              float* __restrict__ z1,
                                              float* __restrict__ sum,
                                              float* __restrict__ sq) {
  __shared__ float ls[64], lq[64];
  int st = blockIdx.x, b = blockIdx.y;
  int lane = threadIdx.x, lm = lane & 15, hl = lane >> 4;
  int scol = st * 16 + lm;
  const float* xb = x + b * 64 * NS;
  v8f acc[4] = {{}, {}, {}, {}};
  for (int ks = 0; ks < 48; ks++) {
    int k0 = ks * 4 + 2 * hl;
    v2f bf;
    {
      int c = k0 / 3, t = k0 % 3, s = scol + t - 1;
      bf.x = (s >= 0 && s < NS) ? xb[c * NS + s] : 0.0f;
    }
    {
      int k1 = k0 + 1, c = k1 / 3, t = k1 % 3, s = scol + t - 1;
      bf.y = (s >= 0 && s < NS) ? xb[c * NS + s] : 0.0f;
    }
    __builtin_prefetch(xb + ((ks * 4 + 4) / 3) * NS + scol, 0, 0);
#pragma unroll
    for (int ot = 0; ot < 4; ot++) {
      int o = ot * 16 + lm;
      v2f af;
      af.x = w1[o * 192 + k0];
      af.y = w1[o * 192 + k0 + 1];
      acc[ot] = WMMA_F32X4(af, bf, acc[ot]);
    }
  }
  for (int i = lane; i < 64; i += 32) { ls[i] = 0.f; lq[i] = 0.f; }
  __syncthreads();
#pragma unroll
  for (int ot = 0; ot < 4; ot++)
#pragma unroll
    for (int r = 0; r < 8; r++) {
      int o = ot * 16 + r + 8 * hl;
      float z = acc[ot][r] + b1[o];
      z1[(b * 64 + o) * NS + scol] = z;
      atomicAdd(&ls[o], z);
      atomicAdd(&lq[o], z * z);
    }
  __syncthreads();
  for (int i = lane; i < 64; i += 32) {
    atomicAdd(&sum[i], ls[i]);
    atomicAdd(&sq[i],  lq[i]);
  }
}

// ---------------------------------------------------------------------------
// kernel 8: fused BN1+ReLU, channel attention (64->4->64 MLP), spatial
// attention (conv1d 2->1,k=7), residual add. One block per batch (L2-resident).
// ---------------------------------------------------------------------------
__global__ __launch_bounds__(256) void k_attn(const float* __restrict__ x,
                                              const float* __restrict__ z1,
                                              const float* __restrict__ s1,
                                              const float* __restrict__ h1,
                                              const float* __restrict__ caw1,
                                              const float* __restrict__ caw2,
                                              const float* __restrict__ saw,
                                              float* __restrict__ zc,
                                              float* __restrict__ out) {
  __shared__ float pS[256], pM[256];
  __shared__ float chA[64], chX[64];
  __shared__ float hid[8];
  __shared__ float ca[64];
  __shared__ float sm[NS], sx[NS];
  int b = blockIdx.x, t = threadIdx.x;
  const float* z1b = z1 + b * 64 * NS;
  const float* xb  = x  + b * 64 * NS;
  float* zcb = zc + b * 64 * NS;
  float* ob  = out + b * 64 * NS;

  // pass 1: per-channel avg & max over S of relu(bn1(z1))
  {
    int o = t >> 2, part = t & 3;
    float sc = s1[o], shf = h1[o];
    float s = 0.f, m = -3.4e38f;
    for (int ss = part * 256; ss < part * 256 + 256; ss++) {
      float v = sc * z1b[o * NS + ss] + shf;
      v = v > 0.f ? v : 0.f;
      s += v; m = v > m ? v : m;
    }
    pS[t] = s; pM[t] = m;
  }
  __syncthreads();
  if (t < 64) {
    float s = 0.f, m = -3.4e38f;
    for (int p = 0; p < 4; p++) {
      s += pS[t * 4 + p];
      float mm = pM[t * 4 + p]; m = mm > m ? mm : m;
    }
    chA[t] = s * (1.0f / NS); chX[t] = m;
  }
  __syncthreads();
  // channel-attention MLP
  if (t < 8) {
    int r = t & 3;
    const float* in = (t < 4) ? chA : chX;
    float h = 0.f;
    for (int c = 0; c < 64; c++) h += caw1[r * 64 + c] * in[c];
    hid[t] = h > 0.f ? h : 0.f;
  }
  __syncthreads();
  if (t < 64) {
    float a = 0.f;
    for (int r = 0; r < 4; r++) a += caw2[t * 4 + r] * (hid[r] + hid[4 + r]);
    ca[t] = 1.0f / (1.0f + expf(-a));
  }
  __syncthreads();
  // pass 2: apply channel attention; per-s mean & max over channels
  for (int ss = t; ss < NS; ss += 256) {
    float s = 0.f, m = -3.4e38f;
    for (int o = 0; o < 64; o++) {
      float v = s1[o] * z1b[o * NS + ss] + h1[o];
      v = v > 0.f ? v : 0.f;
      v *= ca[o];
      zcb[o * NS + ss] = v;
      s += v; m = v > m ? v : m;
    }
    sm[ss] = s * (1.0f / 64.0f); sx[ss] = m;
  }
  __syncthreads();
  // pass 3: spatial conv (2->1, k=7, pad=3) + sigmoid + residual
  for (int ss = t; ss < NS; ss += 256) {
    float a = 0.f;
    for (int j = 0; j < 7; j++) {
      int p = ss + j - 3;
      if (p >= 0 && p < NS) a += saw[j] * sm[p] + saw[7 + j] * sx[p];
    }
    float sa = 1.0f / (1.0f + expf(-a));
    for (int o = 0; o < 64; o++)
      ob[o * NS + ss] = xb[o * NS + ss] + zcb[o * NS + ss] * sa;
  }
}

// ---------------------------------------------------------------------------
extern "C" void kernel_launch(void* const* d_in, const int* in_sizes, int n_in,
                              void* d_out, int out_size, void* d_ws, size_t ws_size,
                              hipStream_t stream) {
  const float* xyz  = (const float*)d_in[0];
  const float* anc  = (const float*)d_in[1];
  const float* alpha = (const float*)d_in[2];
  const float* beta  = (const float*)d_in[3];
  const float* w2  = (const float*)d_in[4];
  const float* b2  = (const float*)d_in[5];
  const float* g2  = (const float*)d_in[6];
  const float* bb2 = (const float*)d_in[7];
  const float* w1  = (const float*)d_in[8];
  const float* b1  = (const float*)d_in[9];
  const float* g1  = (const float*)d_in[10];
  const float* bb1 = (const float*)d_in[11];
  const float* caw1 = (const float*)d_in[12];
  const float* caw2 = (const float*)d_in[13];
  const float* saw  = (const float*)d_in[14];
  float* out = (float*)d_out;

  // workspace layout (floats)
  float* ws = (float*)d_ws;
  float* grp    = ws;                      // 1179648
  float* ctr    = grp + 1179648;           // 1179648
  float* sp     = ctr + 1179648;           // 1024  per-(b,stile) sumsq partials
  float* invstd = sp + 1024;               // 16
  float* bn2sum = invstd + 16;             // 64
  float* bn2sq  = bn2sum + 64;             // 64
  float* bn2sc  = bn2sq + 64;              // 64
  float* bn2sh  = bn2sc + 64;              // 64
  float* xbuf   = bn2sh + 64;              // 1048576  x[B,64,S]
  float* z1     = xbuf + 1048576;          // 1048576  pre-BN conv1 out
  float* zcb    = z1 + 1048576;            // 1048576  channel-attended z
  float* bn1sum = zcb + 1048576;           // 64
  float* bn1sq  = bn1sum + 64;             // 64
  float* bn1sc  = bn1sq + 64;              // 64
  float* bn1sh  = bn1sc + 64;              // 64

  k_zero<<<dim3(4), dim3(256), 0, stream>>>(sp, 1024, bn2sum, 128, bn1sum, 128);
  k_knn<<<dim3(64, 16), dim3(32), 0, stream>>>(xyz, anc, grp, ctr, sp);
  k_std<<<dim3(1), dim3(32), 0, stream>>>(sp, invstd);
  k_c2stats<<<dim3(96, 16), dim3(32), 0, stream>>>(grp, ctr, invstd, alpha, beta,
                                                   w2, b2, bn2sum, bn2sq);
  k_bnfin<<<dim3(1), dim3(64), 0, stream>>>(bn2sum, bn2sq, g2, bb2, bn2sc, bn2sh,
                                            (float)(NB * NS * NK));
  k_c2apply<<<dim3(32, 16), dim3(128), 0, stream>>>(grp, ctr, invstd, alpha, beta,
                                                    w2, b2, bn2sc, bn2sh, xbuf);
  k_conv1<<<dim3(64, 16), dim3(32), 0, stream>>>(xbuf, w1, b1, z1, bn1sum, bn1sq);
  k_bnfin<<<dim3(1), dim3(64), 0, stream>>>(bn1sum, bn1sq, g1, bb1, bn1sc, bn1sh,
                                            (float)(NB * NS));
  k_attn<<<dim3(16), dim3(256), 0, stream>>>(xbuf, z1, bn1sc, bn1sh, caw1, caw2,
                                             saw, zcb, out);
}